// RDFLoss_32882269618684
// MI455X (gfx1250) — compile-verified
//
#include <hip/hip_runtime.h>

typedef __attribute__((ext_vector_type(2))) float v2f;
typedef __attribute__((ext_vector_type(8))) float v8f;

#define NPTS   12288
#define TILE   16
#define NTILE  (NPTS / TILE)   // 768
#define HALFT  (NTILE / 2)     // 384
#define NBINS  10
#define WPB    8               // waves per block
#define FSTRIDE (3 * TILE * WPB)   // floats between a wave's consecutive tiles

// ---------------------------------------------------------------------------
__global__ void rdf_zero_kernel(unsigned int* __restrict__ gacc) {
    if (threadIdx.x < NBINS) gacc[threadIdx.x] = 0u;
}

// ---------------------------------------------------------------------------
// One 16x16 tile: WMMA Gram tile, d^2 reconstruction, cumulative bin counters.
// Ballots for adjacent element pairs are fused into one 64-bit popcount
// (s_bcnt1_i32_b64) to halve scalar-pipe traffic.
// ---------------------------------------------------------------------------
__device__ __forceinline__ void tile_accum(const v2f A, const float rsq[8],
                                           float bx, float by, float bz,
                                           int half, unsigned int (&acc)[NBINS]) {
    // B operand (4x16 f32): lanes 0-15: K0(x),K1(y); lanes 16-31: K2(z),K3(0)
    v2f B;
    B.x = half ? bz : bx;
    B.y = half ? 0.0f : by;
    const float csq = bx * bx + by * by + bz * bz;   // this lane's column ||q||^2

    v8f C = {};
    C = __builtin_amdgcn_wmma_f32_16x16x4_f32(
        /*neg_a=*/false, A, /*neg_b=*/false, B,
        /*c_mod=*/(short)0, C, /*reuse_a=*/false, /*reuse_b=*/false);

    float d2[8];
#pragma unroll
    for (int v = 0; v < 8; ++v)
        d2[v] = __builtin_fmaf(-2.0f, C[v], rsq[v] + csq);
    // d2 < 0 (diagonal fp noise) fails all compares -> bin 0, as reference.

#pragma unroll
    for (int k = 0; k < 9; ++k) {
        const float e2 = (float)((k + 1) * (k + 1));
        unsigned int s = 0;
#pragma unroll
        for (int v = 0; v < 8; v += 2) {
            const unsigned long long m0 = __builtin_amdgcn_ballot_w32(d2[v]     >= e2);
            const unsigned long long m1 = __builtin_amdgcn_ballot_w32(d2[v + 1] >= e2);
            s += (unsigned int)__builtin_popcountll((m1 << 32) | m0);
        }
        acc[k] += s;
    }
    {
        unsigned int s = 0;
#pragma unroll
        for (int v = 0; v < 8; v += 2) {
            const unsigned long long m0 = __builtin_amdgcn_ballot_w32(d2[v]     > 100.0f);
            const unsigned long long m1 = __builtin_amdgcn_ballot_w32(d2[v + 1] > 100.0f);
            s += (unsigned int)__builtin_popcountll((m1 << 32) | m0);
        }
        acc[9] += s;   // strictly beyond r_max
    }
}

// ---------------------------------------------------------------------------
// Run `iters` tiles starting at column-tile jt0, advancing jt by WPB each
// iteration (no wrap inside a phase). Double-buffered column loads.
// ---------------------------------------------------------------------------
__device__ __forceinline__ void run_phase(const float* __restrict__ pc,
                                          int jt0, int iters,
                                          const v2f A, const float rsq[8],
                                          int half, int l16,
                                          unsigned int (&acc)[NBINS]) {
    if (iters <= 0) return;                       // wave-uniform
    const float* p = pc + 3 * (jt0 * TILE + l16); // lane's column point
    float bx = p[0], by = p[1], bz = p[2];
    for (int n = 1; n < iters; ++n) {             // wave-uniform trip count
        p += FSTRIDE;                             // constant 1536 B bump
        const float nx = p[0], ny = p[1], nz = p[2];  // prefetch next tile
        tile_accum(A, rsq, bx, by, bz, half, acc);    // overlaps load latency
        bx = nx; by = ny; bz = nz;
    }
    tile_accum(A, rsq, bx, by, bz, half, acc);
}

// ---------------------------------------------------------------------------
// Symmetric wrap decomposition: block it handles tile offsets
// s=0 (w=1), s=1..383 (w=2), s=384 (w=1): exactly 385 tiles, fully balanced.
// ---------------------------------------------------------------------------
__global__ __launch_bounds__(256)
void rdf_pair_kernel(const float* __restrict__ pc, unsigned int* __restrict__ gacc) {
    __shared__ float        rowsq[TILE];
    __shared__ unsigned int sacc[NBINS];

    const int tid  = threadIdx.x;
    const int lane = tid & 31;
    const int wave = tid >> 5;        // wave-uniform at runtime
    const int half = lane >> 4;
    const int l16  = lane & 15;

    if (tid < NBINS) sacc[tid] = 0u;

    const int it = blockIdx.x;
    const int I0 = it * TILE;

    // A operand (16x4 f32): lanes 0-15: K0(x),K1(y); lanes 16-31: K2(z),K3(0)
    const int arow = I0 + l16;
    const float ax = pc[3 * arow + 0];
    const float ay = pc[3 * arow + 1];
    const float az = pc[3 * arow + 2];
    v2f A;
    A.x = half ? az : ax;
    A.y = half ? 0.0f : ay;

    const float my_rsq = ax * ax + ay * ay + az * az;
    if (tid < TILE) rowsq[tid] = my_rsq;
    __syncthreads();

    float rsq[8];                      // C/D layout: lane<16 -> M=v, else M=v+8
#pragma unroll
    for (int v = 0; v < 8; ++v) rsq[v] = rowsq[half * 8 + v];

    unsigned int acc1[NBINS];          // weight-1 tiles (s=0, s=384)
    unsigned int acc2[NBINS];          // weight-2 tiles (s=1..383)
#pragma unroll
    for (int k = 0; k < NBINS; ++k) { acc1[k] = 0u; acc2[k] = 0u; }

    // ---- Weight-1 tiles: wave 0 -> s=0 (diagonal), wave 1 -> s=384.
    if (wave < 2) {                    // uniform per wave: EXEC all-ones inside
        int jt = (wave == 0) ? it : it + HALFT;
        if (jt >= NTILE) jt -= NTILE;
        const float* p = pc + 3 * (jt * TILE + l16);
        tile_accum(A, rsq, p[0], p[1], p[2], half, acc1);
    }

    // ---- Weight-2 tiles: wave w covers s = w, w+8, ... within [1, 383],
    //      split at the wrap point so each phase is a constant-stride walk.
    {
        const int sBeg  = (wave == 0) ? WPB : wave;
        const int lim   = NTILE - it;                    // first wrapped s
        const int sEndA = (lim < HALFT) ? lim : HALFT;
        const int itersA = (sEndA > sBeg) ? ((sEndA - sBeg + 7) >> 3) : 0;
        const int sB0    = sBeg + itersA * WPB;          // next s in sequence
        const int itersB = (HALFT > sB0) ? ((HALFT - sB0 + 7) >> 3) : 0;

        run_phase(pc, it + sBeg,          itersA, A, rsq, half, l16, acc2);
        run_phase(pc, it + sB0 - NTILE,   itersB, A, rsq, half, l16, acc2);
    }

    // ---- acc are wave-uniform: lane 0 adds wave totals to LDS, then global.
    if (lane == 0) {
#pragma unroll
        for (int k = 0; k < NBINS; ++k)
            atomicAdd(&sacc[k], acc1[k] + 2u * acc2[k]);
    }
    __syncthreads();
    if (tid < NBINS) atomicAdd(&gacc[tid], sacc[tid]);
}

// ---------------------------------------------------------------------------
__global__ void rdf_finalize_kernel(const unsigned int* __restrict__ gacc,
                                    const float* __restrict__ target,
                                    float* __restrict__ out) {
    if (threadIdx.x != 0 || blockIdx.x != 0) return;

    const float total = (float)NPTS * (float)NPTS;    // 9*2^24, exact in f32
    float hist[NBINS];
    float prev = total;                               // s_0 = all ordered pairs
#pragma unroll
    for (int k = 0; k < 9; ++k) {
        const float cur = (float)gacc[k];             // s_{k+1}
        hist[k] = prev - cur;
        prev = cur;
    }
    hist[9] = prev - (float)gacc[9];                  // s_9 - #{d > r_max}

    const float pi      = 3.14159265358979323846f;
    const float r_max   = 10.0f;
    const float dr      = 1.0f;
    const float density = (float)NPTS / (4.0f / 3.0f * pi * r_max * r_max * r_max);

    float loss = 0.0f;
#pragma unroll
    for (int k = 0; k < NBINS; ++k) {
        const float rmid  = ((float)k + 0.5f) * dr;
        const float ideal = 4.0f * pi * rmid * rmid * dr * density * (float)NPTS;
        const float rdf   = hist[k] / ideal;          // ideal > 0, never NaN
        const float diff  = rdf - target[k];
        loss = __builtin_fmaf(diff, diff, loss);
    }
    out[0] = loss / (float)NBINS;
}

// ---------------------------------------------------------------------------
extern "C" void kernel_launch(void* const* d_in, const int* in_sizes, int n_in,
                              void* d_out, int out_size, void* d_ws, size_t ws_size,
                              hipStream_t stream) {
    (void)in_sizes; (void)n_in; (void)out_size; (void)ws_size;
    const float* pc     = (const float*)d_in[0];   // [12288,3] f32
    const float* target = (const float*)d_in[1];   // [10] f32
    unsigned int* gacc  = (unsigned int*)d_ws;     // 10 counters
    float* out          = (float*)d_out;

    rdf_zero_kernel<<<1, 32, 0, stream>>>(gacc);
    rdf_pair_kernel<<<NTILE, 256, 0, stream>>>(pc, gacc);
    rdf_finalize_kernel<<<1, 32, 0, stream>>>(gacc, target, out);
}